// AEVComputer_56341380989216
// MI455X (gfx1250) — compile-verified
//
#include <hip/hip_runtime.h>
#include <math.h>

// ---------------- problem constants ----------------
#define NCONF      16
#define NATOM      64
#define NPAIR      2016          // 64*63/2
#define NSPEC      4
#define NSPPAIR    10
#define RCR_C      5.2f
#define RCA_C      3.5f
#define PI_OVER_RCR 0.604152433f // pi/5.2
#define PI_OVER_RCA 0.897597901f // pi/3.5
#define OUT_PER_ATOM 384         // 64 radial + 320 angular

typedef float v2f __attribute__((ext_vector_type(2)));
typedef float v8f __attribute__((ext_vector_type(8)));

// cos / sin of SHF_Z[k] = (2k+1) * pi/16
__device__ __constant__ float c_cosZ[8] = {
     0.980785280f,  0.831469612f,  0.555570233f,  0.195090322f,
    -0.195090322f, -0.555570233f, -0.831469612f, -0.980785280f };
__device__ __constant__ float c_sinZ[8] = {
     0.195090322f,  0.555570233f,  0.831469612f,  0.980785280f,
     0.980785280f,  0.831469612f,  0.555570233f,  0.195090322f };
__device__ __constant__ float c_shA[4] = { 0.9f, 1.55f, 2.2f, 2.85f };

// fast HW transcendentals (1-ulp class): v_sqrt_f32 / v_rcp_f32, no IEEE fixup chain
__device__ __forceinline__ float fsqrt(float x) { return __builtin_amdgcn_sqrtf(x); }
__device__ __forceinline__ float frcp (float x) { return __builtin_amdgcn_rcpf(x); }

// pair-type id for clipped species (a,b in [0,3]); matches PAIR_TABLE ordering
__device__ __forceinline__ int pairId(int a, int b) {
    a = min(max(a, 0), 3);
    b = min(max(b, 0), 3);
    int lo = min(a, b), hi = max(a, b);
    return lo * 4 - (lo * (lo - 1)) / 2 + (hi - lo);
}

// decode flat triu(k=1) pair index p (n=64) -> (row, col)
__device__ __forceinline__ void decodePair(int p, int& row, int& col) {
    float disc = 16129.0f - 8.0f * (float)p;          // 127^2 - 8p
    int r = (int)floorf((127.0f - fsqrt(disc)) * 0.5f);
    if (r < 0) r = 0;
    while (((r + 1) * (127 - (r + 1))) / 2 <= p) ++r; // f(r) = pairs before row r
    while ((r * (127 - r)) / 2 > p) --r;
    row = r;
    col = p - (r * (127 - r)) / 2 + r + 1;
}

// LDS byte offset of a __shared__ object: low 32 bits of its generic address
__device__ __forceinline__ unsigned ldsOff(const void* p) {
    return (unsigned)(unsigned long long)p;
}

__global__ __launch_bounds__(256) void aev_kernel(
        const int* __restrict__ species,
        const float* __restrict__ coords,
        float* __restrict__ out) {
    __shared__ float  sX[NATOM], sY[NATOM], sZ[NATOM];
    __shared__ int    sSp[NATOM];
    __shared__ float4 stP[8][32];   // {cos_t, sin_t, dm, 2*fc1*fc2*valid}
    __shared__ int    stT[8][32];   // species-pair id

    const int tid  = threadIdx.x;
    const int w    = tid >> 5;          // wave id within block (atom sub-id)
    const int lane = tid & 31;
    const int c    = blockIdx.x >> 3;   // conformation
    const int i    = ((blockIdx.x & 7) << 3) + w;   // my atom

    // --- CDNA5 async memory->LDS copy of the conformation's atoms ---
    if (tid < NATOM) {
        const int g = c * NATOM + tid;
        unsigned lX = ldsOff(&sX[tid]);
        unsigned lY = ldsOff(&sY[tid]);
        unsigned lZ = ldsOff(&sZ[tid]);
        unsigned lS = ldsOff(&sSp[tid]);
        unsigned gc = (unsigned)(g * 12);   // byte offset into coords
        unsigned gs = (unsigned)(g * 4);    // byte offset into species
        asm volatile(
            "global_load_async_to_lds_b32 %0, %4, %8\n\t"
            "global_load_async_to_lds_b32 %1, %5, %8\n\t"
            "global_load_async_to_lds_b32 %2, %6, %8\n\t"
            "global_load_async_to_lds_b32 %3, %7, %9"
            :
            : "v"(lX), "v"(lY), "v"(lZ), "v"(lS),
              "v"(gc), "v"(gc + 4u), "v"(gc + 8u), "v"(gs),
              "s"(coords), "s"(species)
            : "memory");
    }
    asm volatile("s_wait_asynccnt 0" ::: "memory");
    __syncthreads();

    const int m  = lane & 15;   // A-matrix row (r or z) AND B/D column (s or t)
    const int hk = lane >> 4;   // which K sub-pair {0,1} vs {2,3}

    const float xi = sX[i], yi = sY[i], zi = sZ[i];

    // ------------------------------------------------------------------
    // Radial AEV:  D[r,s] += sum_j rt(j,r) * onehot(j,s)  via f32 WMMA K=4
    // ------------------------------------------------------------------
    const float shfR = 0.9f + 0.26875f * (float)m;
    v8f accR = {0.f,0.f,0.f,0.f,0.f,0.f,0.f,0.f};
    #pragma unroll
    for (int j0 = 0; j0 < NATOM; j0 += 4) {
        const int ja = j0 + 2 * hk;
        const int jb = ja + 1;
        v2f Av, Bv;
        {
            float dx = xi - sX[ja], dy = yi - sY[ja], dz = zi - sZ[ja];
            float d  = fsqrt(dx*dx + dy*dy + dz*dz);
            d = (ja == i) ? (RCR_C + 1.0f) : d;
            float fc = (d <= RCR_C) ? fmaf(0.5f, __cosf(PI_OVER_RCR * d), 0.5f) : 0.0f;
            float dd = d - shfR;
            Av.x = 0.25f * __expf(-16.0f * dd * dd) * fc;
            Bv.x = (sSp[ja] == m) ? 1.0f : 0.0f;
        }
        {
            float dx = xi - sX[jb], dy = yi - sY[jb], dz = zi - sZ[jb];
            float d  = fsqrt(dx*dx + dy*dy + dz*dz);
            d = (jb == i) ? (RCR_C + 1.0f) : d;
            float fc = (d <= RCR_C) ? fmaf(0.5f, __cosf(PI_OVER_RCR * d), 0.5f) : 0.0f;
            float dd = d - shfR;
            Av.y = 0.25f * __expf(-16.0f * dd * dd) * fc;
            Bv.y = (sSp[jb] == m) ? 1.0f : 0.0f;
        }
        accR = __builtin_amdgcn_wmma_f32_16x16x4_f32(
            false, Av, false, Bv, (short)0, accR, false, false);
    }

    float* obase = out + (size_t)(c * NATOM + i) * OUT_PER_ATOM;
    if (m < NSPEC) {
        #pragma unroll
        for (int v = 0; v < 8; ++v) {
            int r = v + 8 * hk;            // D: M = vgpr + 8*(lane>=16)
            obase[m * 16 + r] = accR[v];
        }
    }

    // ------------------------------------------------------------------
    // Angular AEV:  D[z,t] += sum_p at(p,z) * onehot_p(p,t)
    // Two 16-row z-halves (z = h*16 + m), K=4 pairs per WMMA.
    // ------------------------------------------------------------------
    const float czv   = c_cosZ[m & 7];        // z & 7 == m & 7 for both halves
    const float szv   = c_sinZ[m & 7];
    const float shaLo = c_shA[m >> 3];        // z>>3 for z = m
    const float shaHi = c_shA[2 + (m >> 3)];  // z>>3 for z = 16 + m

    v8f accLo = {0.f,0.f,0.f,0.f,0.f,0.f,0.f,0.f};
    v8f accHi = {0.f,0.f,0.f,0.f,0.f,0.f,0.f,0.f};

    for (int chunk = 0; chunk < 63; ++chunk) {
        // ---- phase 1: one pair per lane -> packed LDS staging ----
        {
            int p = chunk * 32 + lane;
            int a1, a2;
            decodePair(p, a1, a2);
            float v1x = xi - sX[a1], v1y = yi - sY[a1], v1z = zi - sZ[a1];
            float v2x = xi - sX[a2], v2y = yi - sY[a2], v2z = zi - sZ[a2];
            bool valid = (a1 != i) && (a2 != i);
            float d1 = fsqrt(v1x*v1x + v1y*v1y + v1z*v1z);
            float d2 = fsqrt(v2x*v2x + v2y*v2y + v2z*v2z);
            d1 = valid ? d1 : (RCA_C + 1.0f);
            d2 = valid ? d2 : (RCA_C + 1.0f);
            float dot = v1x*v2x + v1y*v2y + v1z*v2z;
            float ca  = (0.95f * dot) * frcp(fmaxf(d1, 1e-8f) * fmaxf(d2, 1e-8f));
            float sa  = fsqrt(fmaxf(0.0f, 1.0f - ca * ca)); // sin(acos(ca)), acos-free
            float fc1 = (d1 <= RCA_C) ? fmaf(0.5f, __cosf(PI_OVER_RCA * d1), 0.5f) : 0.0f;
            float fc2 = (d2 <= RCA_C) ? fmaf(0.5f, __cosf(PI_OVER_RCA * d2), 0.5f) : 0.0f;
            float4 pk;
            pk.x = ca;
            pk.y = sa;
            pk.z = 0.5f * (d1 + d2);
            pk.w = valid ? (2.0f * fc1 * fc2) : 0.0f;
            stP[w][lane] = pk;                      // one ds_store_b128
            stT[w][lane] = pairId(sSp[a1], sSp[a2]);
        }
        asm volatile("s_wait_dscnt 0" ::: "memory"); // intra-wave LDS handoff

        // ---- phase 2: 8 K-groups of 4 pairs, two z-half WMMAs each ----
        #pragma unroll
        for (int g = 0; g < 8; ++g) {
            const int s0 = g * 4 + 2 * hk;
            const float4 p0 = stP[w][s0];           // ds_load_b128 (broadcast)
            const float4 p1 = stP[w][s0 + 1];
            const int    t0 = stT[w][s0];
            const int    t1 = stT[w][s0 + 1];

            v2f Bv;
            Bv.x = (t0 == m) ? 1.0f : 0.0f;
            Bv.y = (t1 == m) ? 1.0f : 0.0f;

            v2f Alo, Ahi;
            {
                float cs = fmaf(p0.x, czv, p0.y * szv); // cos(theta - shf_z)
                float tt = fmaf(0.5f, cs, 0.5f);
                float t2 = tt*tt, t4 = t2*t2, t8 = t4*t4, t16 = t8*t8;
                float t32 = t16 * t16;                  // ^zeta, shared by halves
                float b  = p0.w * t32;
                float dd = p0.z - shaLo;
                Alo.x = b * __expf(-8.0f * dd * dd);
                dd = p0.z - shaHi;
                Ahi.x = b * __expf(-8.0f * dd * dd);
            }
            {
                float cs = fmaf(p1.x, czv, p1.y * szv);
                float tt = fmaf(0.5f, cs, 0.5f);
                float t2 = tt*tt, t4 = t2*t2, t8 = t4*t4, t16 = t8*t8;
                float t32 = t16 * t16;
                float b  = p1.w * t32;
                float dd = p1.z - shaLo;
                Alo.y = b * __expf(-8.0f * dd * dd);
                dd = p1.z - shaHi;
                Ahi.y = b * __expf(-8.0f * dd * dd);
            }
            accLo = __builtin_amdgcn_wmma_f32_16x16x4_f32(
                false, Alo, false, Bv, (short)0, accLo, false, false);
            accHi = __builtin_amdgcn_wmma_f32_16x16x4_f32(
                false, Ahi, false, Bv, (short)0, accHi, false, false);
        }
        asm volatile("s_wait_dscnt 0" ::: "memory"); // WAR fence before next chunk
    }

    if (m < NSPPAIR) {
        #pragma unroll
        for (int v = 0; v < 8; ++v) {
            int z = v + 8 * hk;
            obase[64 + m * 32 + z]       = accLo[v];   // z in [0,16)
            obase[64 + m * 32 + 16 + z]  = accHi[v];   // z in [16,32)
        }
    }
}

extern "C" void kernel_launch(void* const* d_in, const int* in_sizes, int n_in,
                              void* d_out, int out_size, void* d_ws, size_t ws_size,
                              hipStream_t stream) {
    (void)in_sizes; (void)n_in; (void)d_ws; (void)ws_size; (void)out_size;
    const int*   species = (const int*)d_in[0];
    const float* coords  = (const float*)d_in[1];
    float*       out     = (float*)d_out;
    // 16 conformations * 8 atom-groups; 8 waves (atoms) per block
    aev_kernel<<<dim3(NCONF * 8), dim3(256), 0, stream>>>(species, coords, out);
}